// GAT_transformer_64433099375063
// MI455X (gfx1250) — compile-verified
//
#include <hip/hip_runtime.h>
#include <math.h>

// ---------------------------------------------------------------------------
// GAT + dense self-attention + GAT pipeline for gfx1250 (MI455X).
// FP32 throughout; matmuls use V_WMMA_F32_16X16X4_F32 (exact fp32 matrix op).
// ---------------------------------------------------------------------------

typedef __attribute__((ext_vector_type(2))) float v2f;
typedef __attribute__((ext_vector_type(8))) float v8f;

#define LN_EPS   1e-5f
#define NEGSLOPE 0.2f

// Monotonic float<->uint mapping so atomicMax(u32) == float max (handles neg, -0).
__device__ __forceinline__ unsigned encF(float f) {
    unsigned b = __float_as_uint(f);
    return (b & 0x80000000u) ? ~b : (b | 0x80000000u);
}
__device__ __forceinline__ float decF(unsigned u) {
    unsigned b = (u & 0x80000000u) ? (u ^ 0x80000000u) : ~u;
    return __uint_as_float(b);
}
#define NEG_INF_ENC 0x007FFFFFu  // encF(-inf)

// ---------------------------------------------------------------------------
// init: lmax = enc(-inf), denom = 0, agg = 0 for both GAT layers
// ---------------------------------------------------------------------------
__global__ void k_init(unsigned* lm1, unsigned* lm2, float* dn1, float* dn2,
                       float* ag1, float* ag2, int n) {
    int i = blockIdx.x * blockDim.x + threadIdx.x;
    if (i < n * 8) { lm1[i] = NEG_INF_ENC; lm2[i] = NEG_INF_ENC; dn1[i] = 0.f; dn2[i] = 0.f; }
    if (i < n * 64) ag1[i] = 0.f;
    if (i < n * 48) ag2[i] = 0.f;
}

// ---------------------------------------------------------------------------
// Generic fp32 GEMM via WMMA 16x16x4. One wave per 16x16 C tile.
//   C[tm*16+.., tn*16+..] (+)= A[M,K] @ B[K,N]   (row-major, given strides)
// Ragged N: column ADDRESS clamped (garbage only lands in unstored C columns,
// since C[:,n] depends only on B[:,n]) -> branch-free inner loop.
// Ragged K: single masked tail iteration (value select, not branch).
// B/C addressed by pointer-walk (no per-load 64-bit index multiplies).
// Two accumulators break the WMMA->WMMA RAW chain on the matrix pipe.
// Requires lda even (call sites: 256/64/8); M multiple of 16.
// ---------------------------------------------------------------------------
__global__ void k_gemm_wmma(const float* __restrict__ A, int lda,
                            const float* __restrict__ B, int ldb,
                            float* __restrict__ C, int ldc,
                            int tilesM, int tilesN, int Nact, int Kact,
                            int accumulate) {
    int wave = threadIdx.x >> 5;
    int lane = threadIdx.x & 31;
    int tile = blockIdx.x * (blockDim.x >> 5) + wave;
    if (tile >= tilesM * tilesN) return;
    int tm = tile / tilesN, tn = tile % tilesN;
    int l16 = lane & 15, half = lane >> 4;
    int arow = tm * 16 + l16;
    int col  = tn * 16 + l16;
    bool colok = col < Nact;
    int colc = colok ? col : 0;          // clamped, always-legal address

    v8f c0, c1;
    for (int g = 0; g < 8; ++g) { c0[g] = 0.f; c1[g] = 0.f; }
    if (accumulate) {
        const float* Cp = C + (size_t)(tm * 16 + 8 * half) * ldc + colc;
        for (int g = 0; g < 8; ++g) { c0[g] = *Cp; Cp += ldc; }  // garbage ok if !colok
    }

    const float* Ap = A + (size_t)arow * lda + 2 * half;
    const float* Bp = B + (size_t)(2 * half) * ldb + colc;   // walks down K
    int Kmain = Kact & ~3;
    int k0 = 0;
    // main: pairs of WMMA steps on independent accumulators
    for (; k0 + 8 <= Kmain; k0 += 8) {
        v2f a0 = *(const v2f*)(Ap + k0);
        v2f a1 = *(const v2f*)(Ap + k0 + 4);
        v2f b0, b1;
        b0.x = Bp[0];
        b0.y = Bp[ldb];
        b1.x = Bp[4 * (size_t)ldb];
        b1.y = Bp[5 * (size_t)ldb];
        Bp += 8 * (size_t)ldb;
        c0 = __builtin_amdgcn_wmma_f32_16x16x4_f32(false, a0, false, b0, (short)0, c0, false, false);
        c1 = __builtin_amdgcn_wmma_f32_16x16x4_f32(false, a1, false, b1, (short)0, c1, false, false);
    }
    for (; k0 < Kmain; k0 += 4) {
        v2f a = *(const v2f*)(Ap + k0);
        v2f b;
        b.x = Bp[0];
        b.y = Bp[ldb];
        Bp += 4 * (size_t)ldb;
        c0 = __builtin_amdgcn_wmma_f32_16x16x4_f32(false, a, false, b, (short)0, c0, false, false);
    }
    if (Kmain < Kact) {                   // ragged-K tail (value-masked)
        int kk = Kmain + 2 * half;
        int ka = (kk     < Kact) ? kk     : Kact - 1;
        int kb = (kk + 1 < Kact) ? kk + 1 : Kact - 1;
        float ax = A[(size_t)arow * lda + ka];
        float ay = A[(size_t)arow * lda + kb];
        float bx = B[(size_t)ka * ldb + colc];
        float by = B[(size_t)kb * ldb + colc];
        v2f a, b;
        a.x = (kk     < Kact) ? ax : 0.f;
        a.y = (kk + 1 < Kact) ? ay : 0.f;
        b.x = (kk     < Kact) ? bx : 0.f;
        b.y = (kk + 1 < Kact) ? by : 0.f;
        c0 = __builtin_amdgcn_wmma_f32_16x16x4_f32(false, a, false, b, (short)0, c0, false, false);
    }

    if (colok) {
        float* Cp = C + (size_t)(tm * 16 + 8 * half) * ldc + col;
        for (int g = 0; g < 8; ++g) { *Cp = c0[g] + c1[g]; Cp += ldc; }
    }
}

// ---------------------------------------------------------------------------
// Flash attention: out[N,6] = softmax(Q@K^T / 8) @ V.  Q,K: [N,64]; V: [N,8] pad.
// One wave per 16 query rows; 16-key tiles; 16 WMMA f32 16x16x4 per tile
// (two interleaved accumulators). C layout: lane = key-col(l16) + 16*(row/8),
// vgpr g = row%8. Each lane accumulates its key's P*V partial; reduced across
// the 16-lane half at the end.
// ---------------------------------------------------------------------------
__global__ void k_flash_attn(const float* __restrict__ Q, const float* __restrict__ K,
                             const float* __restrict__ V, float* __restrict__ O,
                             int n) {
    int wave = threadIdx.x >> 5, lane = threadIdx.x & 31;
    int l16 = lane & 15, half = lane >> 4;
    int qrow0 = (blockIdx.x * 4 + wave) * 16;

    v2f a[16];  // Q tile 16x64 in A-fragments, pre-scaled by 1/sqrt(64)
    {
        const float* qp = Q + (size_t)(qrow0 + l16) * 64 + 2 * half;
        for (int t = 0; t < 16; ++t) {
            v2f qv = *(const v2f*)(qp + 4 * t);
            a[t].x = qv.x * 0.125f;
            a[t].y = qv.y * 0.125f;
        }
    }

    float m[8], s[8], acc[8][6];
    for (int g = 0; g < 8; ++g) {
        m[g] = -INFINITY; s[g] = 0.f;
        for (int c = 0; c < 6; ++c) acc[g][c] = 0.f;
    }

    const float* kp = K + (size_t)l16 * 64 + 2 * half;   // walks down keys
    const float* vp = V + (size_t)l16 * 8;
    for (int key0 = 0; key0 < n; key0 += 16, kp += 16 * 64, vp += 16 * 8) {
        v2f b[16];
        for (int t = 0; t < 16; ++t) b[t] = *(const v2f*)(kp + 4 * t);

        v8f s0, s1;
        for (int g = 0; g < 8; ++g) { s0[g] = 0.f; s1[g] = 0.f; }
        for (int t = 0; t < 16; t += 2) {
            s0 = __builtin_amdgcn_wmma_f32_16x16x4_f32(false, a[t],     false, b[t],     (short)0, s0, false, false);
            s1 = __builtin_amdgcn_wmma_f32_16x16x4_f32(false, a[t + 1], false, b[t + 1], (short)0, s1, false, false);
        }

        float vv[6];
        for (int c = 0; c < 6; ++c) vv[c] = vp[c];

        for (int g = 0; g < 8; ++g) {
            float x = s0[g] + s1[g];
            float rm = x;
            for (int off = 1; off < 16; off <<= 1) rm = fmaxf(rm, __shfl_xor(rm, off));
            float nm = fmaxf(m[g], rm);
            float corr = __expf(m[g] - nm);     // first tile: exp(-inf)=0
            float p = __expf(x - nm);
            float ps = p;
            for (int off = 1; off < 16; off <<= 1) ps += __shfl_xor(ps, off);
            s[g] = s[g] * corr + ps;
            m[g] = nm;
            for (int c = 0; c < 6; ++c) acc[g][c] = acc[g][c] * corr + p * vv[c];
        }
    }

    for (int g = 0; g < 8; ++g) {
        float inv = 1.f / s[g];
        for (int c = 0; c < 6; ++c) {
            float t = acc[g][c];
            for (int off = 1; off < 16; off <<= 1) t += __shfl_xor(t, off);
            if (l16 == c) O[(size_t)(qrow0 + g + 8 * half) * 8 + c] = t * inv;
        }
    }
}

// ---------------------------------------------------------------------------
// asrc[n,h] = sum_c h[n, h*C+c]*a_src[h,c]; adst likewise. One thread/(node,head).
// ---------------------------------------------------------------------------
__global__ void k_attn_coef(const float* __restrict__ h, int stride, int C,
                            const float* __restrict__ a_src, const float* __restrict__ a_dst,
                            float* __restrict__ osrc, float* __restrict__ odst, int n) {
    int i = blockIdx.x * blockDim.x + threadIdx.x;
    if (i >= n * 8) return;
    int nd = i >> 3, hh = i & 7;
    const float* hp = h + (size_t)nd * stride + hh * C;
    float s = 0.f, d = 0.f;
    for (int c = 0; c < C; ++c) { s += hp[c] * a_src[hh * C + c]; d += hp[c] * a_dst[hh * C + c]; }
    osrc[i] = s; odst[i] = d;
}

// segment max of leaky_relu(asrc[src]+adst[dst]) over dst; self-loops appended.
__global__ void k_edge_max(const int* __restrict__ ei, int E, int n,
                           const float* __restrict__ asrc, const float* __restrict__ adst,
                           unsigned* __restrict__ lmax) {
    int i = blockIdx.x * blockDim.x + threadIdx.x;
    if (i >= (E + n) * 8) return;
    int e = i >> 3, hh = i & 7;
    int src, dst;
    if (e < E) { src = ei[e]; dst = ei[E + e]; } else { src = dst = e - E; }
    float lg = asrc[src * 8 + hh] + adst[dst * 8 + hh];
    lg = (lg >= 0.f) ? lg : NEGSLOPE * lg;
    atomicMax(&lmax[dst * 8 + hh], encF(lg));
}

// ex = exp(logit - max); denom[dst,h] += ex; agg[dst, h*C..] += ex * feat[src, h*C..]
__global__ void k_edge_acc(const int* __restrict__ ei, int E, int n,
                           const float* __restrict__ asrc, const float* __restrict__ adst,
                           const unsigned* __restrict__ lmax,
                           const float* __restrict__ feat, int stride, int C,
                           float* __restrict__ denom, float* __restrict__ agg) {
    int i = blockIdx.x * blockDim.x + threadIdx.x;
    if (i >= (E + n) * 8) return;
    int e = i >> 3, hh = i & 7;
    int src, dst;
    if (e < E) { src = ei[e]; dst = ei[E + e]; } else { src = dst = e - E; }
    float lg = asrc[src * 8 + hh] + adst[dst * 8 + hh];
    lg = (lg >= 0.f) ? lg : NEGSLOPE * lg;
    float ex = __expf(lg - decF(lmax[dst * 8 + hh]));
    atomicAdd(&denom[dst * 8 + hh], ex);
    const float* fp = feat + (size_t)src * stride + hh * C;
    float* ap = agg + (size_t)dst * stride + hh * C;   // stride == 8*C both layers
    for (int c = 0; c < C; ++c) atomicAdd(&ap[c], ex * fp[c]);
}

// out1/denom + b1 -> LayerNorm(64) -> ELU
__global__ void k_node_fin1(const float* __restrict__ agg, const float* __restrict__ denom,
                            const float* __restrict__ b1,
                            const float* __restrict__ lng, const float* __restrict__ lnb,
                            float* __restrict__ h, int n) {
    int nd = blockIdx.x * blockDim.x + threadIdx.x;
    if (nd >= n) return;
    float t[64], mu = 0.f;
    for (int i = 0; i < 64; ++i) {
        t[i] = agg[(size_t)nd * 64 + i] / denom[nd * 8 + (i >> 3)] + b1[i];
        mu += t[i];
    }
    mu *= (1.f / 64.f);
    float var = 0.f;
    for (int i = 0; i < 64; ++i) { float d = t[i] - mu; var += d * d; }
    var *= (1.f / 64.f);
    float r = rsqrtf(var + LN_EPS);
    for (int i = 0; i < 64; ++i) {
        float y = (t[i] - mu) * r * lng[i] + lnb[i];
        h[(size_t)nd * 64 + i] = (y > 0.f) ? y : expm1f(y);   // ELU(alpha=1)
    }
}

// LayerNorm over 6 attention outputs; write padded [N,8] with zeros at 6,7.
__global__ void k_ln2(const float* __restrict__ ht, const float* __restrict__ g,
                      const float* __restrict__ b, float* __restrict__ o, int n) {
    int nd = blockIdx.x * blockDim.x + threadIdx.x;
    if (nd >= n) return;
    float t[6], mu = 0.f;
    for (int c = 0; c < 6; ++c) { t[c] = ht[(size_t)nd * 8 + c]; mu += t[c]; }
    mu *= (1.f / 6.f);
    float var = 0.f;
    for (int c = 0; c < 6; ++c) { float d = t[c] - mu; var += d * d; }
    var *= (1.f / 6.f);
    float r = rsqrtf(var + LN_EPS);
    for (int c = 0; c < 6; ++c) o[(size_t)nd * 8 + c] = (t[c] - mu) * r * g[c] + b[c];
    o[(size_t)nd * 8 + 6] = 0.f;
    o[(size_t)nd * 8 + 7] = 0.f;
}

// mean over 8 heads of agg2/denom2, + b2, log_softmax over 6 classes.
__global__ void k_final(const float* __restrict__ agg2, const float* __restrict__ denom2,
                        const float* __restrict__ b2, float* __restrict__ out, int n) {
    int nd = blockIdx.x * blockDim.x + threadIdx.x;
    if (nd >= n) return;
    float v[6];
    for (int c = 0; c < 6; ++c) {
        float acc = 0.f;
        for (int hh = 0; hh < 8; ++hh)
            acc += agg2[(size_t)nd * 48 + hh * 6 + c] / denom2[nd * 8 + hh];
        v[c] = acc * 0.125f + b2[c];
    }
    float mx = v[0];
    for (int c = 1; c < 6; ++c) mx = fmaxf(mx, v[c]);
    float se = 0.f;
    for (int c = 0; c < 6; ++c) se += __expf(v[c] - mx);
    float lse = logf(se);
    for (int c = 0; c < 6; ++c) out[(size_t)nd * 6 + c] = v[c] - mx - lse;
}

// ---------------------------------------------------------------------------
static inline void launch_gemm(const float* A, int lda, const float* B, int ldb,
                               float* C, int ldc, int tilesM, int tilesN,
                               int Nact, int Kact, int accum, hipStream_t s) {
    int tiles = tilesM * tilesN;
    k_gemm_wmma<<<(tiles + 3) / 4, 128, 0, s>>>(A, lda, B, ldb, C, ldc,
                                                tilesM, tilesN, Nact, Kact, accum);
}

extern "C" void kernel_launch(void* const* d_in, const int* in_sizes, int n_in,
                              void* d_out, int out_size, void* d_ws, size_t ws_size,
                              hipStream_t stream) {
    (void)n_in; (void)out_size; (void)ws_size;
    const float* x      = (const float*)d_in[0];
    const int*   ei     = (const int*)d_in[1];     // int32 (JAX x64 disabled)
    const float* W1     = (const float*)d_in[2];
    const float* a_src1 = (const float*)d_in[3];
    const float* a_dst1 = (const float*)d_in[4];
    const float* b1     = (const float*)d_in[5];
    const float* ln1_g  = (const float*)d_in[6];
    const float* ln1_b  = (const float*)d_in[7];
    const float* Wq     = (const float*)d_in[8];
    const float* Wk     = (const float*)d_in[9];
    const float* Wv     = (const float*)d_in[10];
    const float* ln2_g  = (const float*)d_in[11];
    const float* ln2_b  = (const float*)d_in[12];
    const float* W2     = (const float*)d_in[13];
    const float* a_src2 = (const float*)d_in[14];
    const float* a_dst2 = (const float*)d_in[15];
    const float* b2     = (const float*)d_in[16];
    float* out = (float*)d_out;

    const int N = in_sizes[0] / 256;   // 8192
    const int E = in_sizes[1] / 2;     // 262144
    const int Etot = E + N;

    float* w = (float*)d_ws;
    size_t o = 0;
    float* h0    = w + o; o += (size_t)N * 64;
    float* h     = w + o; o += (size_t)N * 64;
    float* q     = w + o; o += (size_t)N * 64;
    float* kmat  = w + o; o += (size_t)N * 64;
    float* v     = w + o; o += (size_t)N * 8;    // padded [N,8], cols 0..5 valid
    float* ht    = w + o; o += (size_t)N * 8;
    float* htln  = w + o; o += (size_t)N * 8;
    float* hpre2 = w + o; o += (size_t)N * 48;
    float* as1   = w + o; o += (size_t)N * 8;
    float* ad1   = w + o; o += (size_t)N * 8;
    float* dn1   = w + o; o += (size_t)N * 8;
    float* ag1   = w + o; o += (size_t)N * 64;
    float* as2   = w + o; o += (size_t)N * 8;
    float* ad2   = w + o; o += (size_t)N * 8;
    float* dn2   = w + o; o += (size_t)N * 8;
    float* ag2   = w + o; o += (size_t)N * 48;
    unsigned* lm1 = (unsigned*)(w + o); o += (size_t)N * 8;
    unsigned* lm2 = (unsigned*)(w + o); o += (size_t)N * 8;

    const int T = 256;
    // 0) init scratch accumulators
    k_init<<<(N * 64 + T - 1) / T, T, 0, stream>>>(lm1, lm2, dn1, dn2, ag1, ag2, N);
    // 1) h0 = x @ W1   [N,256]x[256,64]
    launch_gemm(x, 256, W1, 64, h0, 64, N / 16, 4, 64, 256, 0, stream);
    // 2) GAT layer 1
    k_attn_coef<<<(N * 8 + T - 1) / T, T, 0, stream>>>(h0, 64, 8, a_src1, a_dst1, as1, ad1, N);
    k_edge_max <<<(Etot * 8 + T - 1) / T, T, 0, stream>>>(ei, E, N, as1, ad1, lm1);
    k_edge_acc <<<(Etot * 8 + T - 1) / T, T, 0, stream>>>(ei, E, N, as1, ad1, lm1, h0, 64, 8, dn1, ag1);
    k_node_fin1<<<(N + T - 1) / T, T, 0, stream>>>(ag1, dn1, b1, ln1_g, ln1_b, h, N);
    // 3) q,k,v projections
    launch_gemm(h, 64, Wq, 64, q,    64, N / 16, 4, 64, 64, 0, stream);
    launch_gemm(h, 64, Wk, 64, kmat, 64, N / 16, 4, 64, 64, 0, stream);
    launch_gemm(h, 64, Wv, 6,  v,    8,  N / 16, 1, 6,  64, 0, stream);
    // 4) flash self-attention -> ht[N,8(6)]
    k_flash_attn<<<N / 64, 128, 0, stream>>>(q, kmat, v, ht, N);
    k_ln2<<<(N + T - 1) / T, T, 0, stream>>>(ht, ln2_g, ln2_b, htln, N);
    // 5) hpre2 = h @ W2[0:64] + htln @ W2[64:70]    [N,48]
    launch_gemm(h,    64, W2,           48, hpre2, 48, N / 16, 3, 48, 64, 0, stream);
    launch_gemm(htln, 8,  W2 + 64 * 48, 48, hpre2, 48, N / 16, 3, 48, 6,  1, stream);
    // 6) GAT layer 2 (C=6 per head, mean over heads)
    k_attn_coef<<<(N * 8 + T - 1) / T, T, 0, stream>>>(hpre2, 48, 6, a_src2, a_dst2, as2, ad2, N);
    k_edge_max <<<(Etot * 8 + T - 1) / T, T, 0, stream>>>(ei, E, N, as2, ad2, lm2);
    k_edge_acc <<<(Etot * 8 + T - 1) / T, T, 0, stream>>>(ei, E, N, as2, ad2, lm2, hpre2, 48, 6, dn2, ag2);
    // 7) head mean + bias + log_softmax
    k_final<<<(N + T - 1) / T, T, 0, stream>>>(ag2, dn2, b2, out, N);
}